// GARCH11_5446018532019
// MI455X (gfx1250) — compile-verified
//
#include <hip/hip_runtime.h>
#include <hip/hip_bf16.h>

// GARCH(1,1) as an affine parallel scan:
//   s_t = a_t*s_{t-1} + OMEGA,  a_t = ALPHA*z_{t-1}^2 + BETA  (z_{-1}:=0, s_{-1}:=1)
//   x_t = sqrt(s_t) * z_t
// Memory floor: 256MB in + 256MB out ~= 22us @ 23.3 TB/s. One pass through HBM
// each way; all rescans hit LDS. Async global<->LDS (ASYNCcnt) for staging.
// Double-buffered: both tiles' async loads issued up front (one wait), tile-0
// stores drain in background while tile-1 computes; single drain at the end.

#define G_OMEGA 0.05f
#define G_ALPHA 0.10f
#define G_BETA  0.85f

constexpr int T        = 4096;
constexpr int D        = 4;
constexpr int THREADS  = 256;
constexpr int TILE_T   = 2048;                 // timesteps per LDS tile
constexpr int NTILES   = T / TILE_T;           // 2
constexpr int NCHUNK   = THREADS / D;          // 64 chunks per tile (per dim)
constexpr int CHUNK_T  = TILE_T / NCHUNK;      // 32 timesteps per thread-chunk
constexpr int V4_PER_THREAD = TILE_T / THREADS; // 8 float4 (one float4 == one timestep)
// Skewed LDS layout: +4 words every 32 timesteps -> conflict-free across chunks,
// still 16B-aligned for B128 async ops.
constexpr int ZS_WORDS = TILE_T * 4 + (TILE_T / 32) * 4; // 8448 floats

__device__ __forceinline__ int zoff(int t) { return t * 4 + ((t >> 5) << 2); }

#define AS_G __attribute__((address_space(1)))
#define AS_L __attribute__((address_space(3)))

typedef int v4i __attribute__((ext_vector_type(4)));

#if defined(__gfx1250__) && \
    __has_builtin(__builtin_amdgcn_global_load_async_to_lds_b128) && \
    __has_builtin(__builtin_amdgcn_global_store_async_from_lds_b128)
#define HAVE_ASYNC_LDS 1
#endif

__device__ __forceinline__ void async_g2l_b128(const float* g, float* l) {
#ifdef HAVE_ASYNC_LDS
  __builtin_amdgcn_global_load_async_to_lds_b128((AS_G v4i*)g, (AS_L v4i*)l, 0, 0);
#else
  const float4 v = *reinterpret_cast<const float4*>(g);
  l[0] = v.x; l[1] = v.y; l[2] = v.z; l[3] = v.w;
#endif
}

__device__ __forceinline__ void async_l2g_b128(float* g, const float* l) {
#ifdef HAVE_ASYNC_LDS
  __builtin_amdgcn_global_store_async_from_lds_b128((AS_G v4i*)g, (AS_L v4i*)l, 0, 0);
#else
  float4 v; v.x = l[0]; v.y = l[1]; v.z = l[2]; v.w = l[3];
  *reinterpret_cast<float4*>(g) = v;
#endif
}

__device__ __forceinline__ void wait_async0() {
#ifdef HAVE_ASYNC_LDS
#if __has_builtin(__builtin_amdgcn_s_wait_asynccnt)
  __builtin_amdgcn_s_wait_asynccnt(0);
#else
  asm volatile("s_wait_asynccnt 0" ::: "memory");
#endif
#endif
}

__global__ __launch_bounds__(THREADS)
void garch11_scan_kernel(const float* __restrict__ z, float* __restrict__ x) {
  __shared__ float zs0[ZS_WORDS];   // tile 0 buffer
  __shared__ float zs1[ZS_WORDS];   // tile 1 buffer
  __shared__ float As[THREADS];
  __shared__ float Bs[THREADS];
  __shared__ float scarry[D];   // sigma2 entering current tile (s_{-1} of tile)
  __shared__ float zcarry[D];   // z at last timestep of previous tile

  const int tid   = threadIdx.x;
  const int dim   = tid & 3;
  const int chunk = tid >> 2;
  const int t0    = chunk * CHUNK_T;
  const long long nbase = (long long)blockIdx.x * (long long)(T * D);

  // ---- issue ALL tile loads up front (async, coalesced 512B/wave/op) ----
  {
    const float* g0 = z + nbase;
    const float* g1 = z + nbase + (long long)(TILE_T * D);
    for (int it = 0; it < V4_PER_THREAD; ++it) {
      const int t = it * THREADS + tid;          // one float4 == z[t, 0..3]
      async_g2l_b128(g0 + t * 4, &zs0[zoff(t)]);
    }
    for (int it = 0; it < V4_PER_THREAD; ++it) {
      const int t = it * THREADS + tid;
      async_g2l_b128(g1 + t * 4, &zs1[zoff(t)]);
    }
  }
  if (tid < D) { scarry[tid] = 1.0f; zcarry[tid] = 0.0f; }  // SIGMA2_0, eps_{-1}=0
  wait_async0();
  __syncthreads();

  for (int tile = 0; tile < NTILES; ++tile) {
    float* zs = tile ? zs1 : zs0;

    // ---- pass 1: per-chunk affine composition (A,B): s_out = A*s_in + B ----
    float zp = (chunk == 0) ? zcarry[dim] : zs[zoff(t0 - 1) + dim];
    float A = 1.0f, B = 0.0f;
    for (int i = 0; i < CHUNK_T; ++i) {
      const float zc = zs[zoff(t0 + i) + dim];
      const float a  = fmaf(G_ALPHA * zp, zp, G_BETA);   // a_t = alpha*z_{t-1}^2 + beta
      A = a * A;
      B = fmaf(a, B, G_OMEGA);
      zp = zc;
    }
    As[tid] = A; Bs[tid] = B;
    __syncthreads();

    // ---- inclusive Hillis–Steele scan over 64 chunks (per dim) ----
    for (int off = 1; off < NCHUNK; off <<= 1) {
      const float a2 = As[tid], b2 = Bs[tid];
      float na = a2, nb = b2;
      if (chunk >= off) {
        const float a1 = As[tid - 4 * off], b1 = Bs[tid - 4 * off];
        na = a2 * a1;
        nb = fmaf(a2, b1, b2);
      }
      __syncthreads();
      As[tid] = na; Bs[tid] = nb;
      __syncthreads();
    }

    // sigma2 entering this chunk = prefix map of chunks [0..chunk-1] applied
    // to the tile-entry value.
    const float sval  = scarry[dim];
    const float s_in  = (chunk == 0) ? sval
                        : fmaf(As[tid - 4], sval, Bs[tid - 4]);
    const float zp0   = (chunk == 0) ? zcarry[dim] : zs[zoff(t0 - 1) + dim];
    const float Afull = As[tid];     // full-tile map held by chunk NCHUNK-1
    const float Bfull = Bs[tid];
    __syncthreads();  // all boundary reads of zs/scarry done before overwrite

    // ---- pass 2: replay chunk, write x in place over z in LDS ----
    float s = s_in;
    zp = zp0;
    for (int i = 0; i < CHUNK_T; ++i) {
      const int idx  = zoff(t0 + i) + dim;
      const float zc = zs[idx];
      const float a  = fmaf(G_ALPHA * zp, zp, G_BETA);
      s = fmaf(a, s, G_OMEGA);
      s = fmaxf(s, 1e-12f);                      // reference clamp (never binds)
      zs[idx] = sqrtf(s) * zc;
      zp = zc;
    }
    if (chunk == NCHUNK - 1) {                   // carry to next tile
      scarry[dim] = fmaf(Afull, sval, Bfull);
      zcarry[dim] = zp;
    }
    __syncthreads();

    // ---- fire async stores for this tile; NO wait — tile 1 computes out of
    //      the other buffer while these drain in the background ----
    float* gdst = x + nbase + (long long)tile * (TILE_T * D);
    for (int it = 0; it < V4_PER_THREAD; ++it) {
      const int t = it * THREADS + tid;
      async_l2g_b128(gdst + t * 4, &zs[zoff(t)]);
    }
  }

  wait_async0();   // drain all outstanding async stores before wave end
}

extern "C" void kernel_launch(void* const* d_in, const int* in_sizes, int n_in,
                              void* d_out, int out_size, void* d_ws, size_t ws_size,
                              hipStream_t stream) {
  const float* z = (const float*)d_in[0];
  float* x = (float*)d_out;
  const int N = in_sizes[0] / (T * D);   // 4096 independent series
  garch11_scan_kernel<<<N, THREADS, 0, stream>>>(z, x);
}